// CTPNLoss_39256001086160
// MI455X (gfx1250) — compile-verified
//
#include <hip/hip_runtime.h>

// Problem constants (match reference)
#define K_ANCH 10
#define H_DIM  512
#define W_DIM  768
#define HW     (H_DIM * W_DIM)
#define NP     64
#define NN     64
#define NV     128
#define NO     64
#define NS     128

typedef __attribute__((ext_vector_type(2))) float v2f;
typedef __attribute__((ext_vector_type(8))) float v8f;

__device__ __forceinline__ float smooth_l1(float d) {
    float a = fabsf(d);
    return (a < 1.0f) ? 0.5f * d * d : (a - 0.5f);
}

// Full-precision FP32 wave32 reduction via V_WMMA_F32_16X16X4_F32.
// A (16x4 f32): lane m holds K=0 (a[0]), K=1 (a[1]); lane m+16 holds K=2,3.
// With B = ones: D[m][n] = a0(m) + a1(m) + a0(m+16) + a1(m+16).
// Setting a1 = 0 gives D[m][*] = p(m) + p(m+16).
// D VGPR r: lanes 0-15 = rows 0..7, lanes 16-31 = rows 8..15, so summing the
// 8 D registers leaves half-sums in the two lane halves; one shfl_xor(16)
// completes the 32-lane total (result in every lane).
__device__ __forceinline__ float wave_sum_wmma(float p) {
    v2f a; a[0] = p;    a[1] = 0.0f;
    v2f b; b[0] = 1.0f; b[1] = 1.0f;
    v8f c = {};
    v8f d = __builtin_amdgcn_wmma_f32_16x16x4_f32(
        /*neg_a=*/false, a, /*neg_b=*/false, b,
        /*c_mod=*/(short)0, c, /*reuse_a=*/false, /*reuse_b=*/false);
    float t = ((d[0] + d[1]) + (d[2] + d[3])) + ((d[4] + d[5]) + (d[6] + d[7]));
    t += __shfl_xor(t, 16, 32);
    return t;
}

__global__ void __launch_bounds__(128)
ctpn_loss_kernel(const float* __restrict__ score,
                 const float* __restrict__ vpred,
                 const float* __restrict__ side,
                 const int* __restrict__ pos_x, const int* __restrict__ pos_y,
                 const int* __restrict__ pos_a,
                 const int* __restrict__ neg_x, const int* __restrict__ neg_y,
                 const int* __restrict__ neg_a,
                 const int* __restrict__ v_x, const int* __restrict__ v_y,
                 const int* __restrict__ v_a,
                 const float* __restrict__ v_target,
                 const int* __restrict__ o_x, const int* __restrict__ o_y,
                 const int* __restrict__ o_a,
                 const float* __restrict__ o_target,
                 float* __restrict__ out) {
    const int tid = threadIdx.x;  // 0..127, 4 waves of 32

    float s_cls = 0.0f, s_v = 0.0f, s_o = 0.0f;

    // ---- classification partial: tid<64 -> pos sample (target=1), else neg ----
    {
        int i = tid;
        const int* gx;
        const int* gy;
        const int* ga;
        int tgt;
        if (i < NP) {
            gx = pos_x; gy = pos_y; ga = pos_a; tgt = 1;
        } else {
            i -= NP;
            gx = neg_x; gy = neg_y; ga = neg_a; tgt = 0;
        }
        int x = gx[i], y = gy[i], aa = ga[i];
        int base = (aa * 2) * HW + y * W_DIM + x;
        float c0 = score[base];
        float c1 = score[base + HW];
        // stable logsumexp over the two logits
        float m   = fmaxf(c0, c1);
        float lse = m + logf(expf(c0 - m) + expf(c1 - m));
        s_cls = lse - (tgt ? c1 : c0);   // -log_softmax[target]
    }

    // ---- vertical regression partial: one sample per thread (NV == 128) ----
    {
        int x = v_x[tid], y = v_y[tid], aa = v_a[tid];
        int base = (aa * 2) * HW + y * W_DIM + x;
        float p0 = vpred[base];
        float p1 = vpred[base + HW];
        s_v = smooth_l1(p0 - v_target[2 * tid + 0]) +
              smooth_l1(p1 - v_target[2 * tid + 1]);
    }

    // ---- side refinement partial: first 64 threads ----
    if (tid < NO) {
        int x = o_x[tid], y = o_y[tid], aa = o_a[tid];
        float p = side[aa * HW + y * W_DIM + x];
        s_o = smooth_l1(p - o_target[tid]);
    }

    // ---- wave-level FP32 WMMA reductions (EXEC is all ones here) ----
    float wcls = wave_sum_wmma(s_cls);
    float wv   = wave_sum_wmma(s_v);
    float wo   = wave_sum_wmma(s_o);

    // ---- cross-wave combine via LDS ----
    __shared__ float red[4][3];
    int wave = tid >> 5;
    if ((tid & 31) == 0) {
        red[wave][0] = wcls;
        red[wave][1] = wv;
        red[wave][2] = wo;
    }
    __syncthreads();

    if (tid == 0) {
        float Sc = (red[0][0] + red[1][0]) + (red[2][0] + red[3][0]);
        float Sv = (red[0][1] + red[1][1]) + (red[2][1] + red[3][1]);
        float So = (red[0][2] + red[1][2]) + (red[2][2] + red[3][2]);
        float cls_loss = Sc / (float)NS;
        float v_loss   = Sv / (float)(NV * 2);
        float o_loss   = So / (float)NO;
        out[0] = cls_loss + v_loss + o_loss;
        out[1] = cls_loss;
        out[2] = v_loss;
        out[3] = o_loss;
    }
}

extern "C" void kernel_launch(void* const* d_in, const int* in_sizes, int n_in,
                              void* d_out, int out_size, void* d_ws, size_t ws_size,
                              hipStream_t stream) {
    (void)in_sizes; (void)n_in; (void)out_size; (void)d_ws; (void)ws_size;

    const float* score  = (const float*)d_in[0];
    const float* vpred  = (const float*)d_in[1];
    const float* side   = (const float*)d_in[2];
    const int* pos_x    = (const int*)d_in[3];
    const int* pos_y    = (const int*)d_in[4];
    const int* pos_a    = (const int*)d_in[5];
    const int* neg_x    = (const int*)d_in[6];
    const int* neg_y    = (const int*)d_in[7];
    const int* neg_a    = (const int*)d_in[8];
    const int* v_x      = (const int*)d_in[9];
    const int* v_y      = (const int*)d_in[10];
    const int* v_a      = (const int*)d_in[11];
    const float* v_tgt  = (const float*)d_in[12];
    const int* o_x      = (const int*)d_in[13];
    const int* o_y      = (const int*)d_in[14];
    const int* o_a      = (const int*)d_in[15];
    const float* o_tgt  = (const float*)d_in[16];
    float* out          = (float*)d_out;

    ctpn_loss_kernel<<<1, 128, 0, stream>>>(score, vpred, side,
                                            pos_x, pos_y, pos_a,
                                            neg_x, neg_y, neg_a,
                                            v_x, v_y, v_a, v_tgt,
                                            o_x, o_y, o_a, o_tgt,
                                            out);
}